// IntSoftmax_14525579395610
// MI455X (gfx1250) — compile-verified
//
#include <hip/hip_runtime.h>
#include <cstdint>

// ---------------------------------------------------------------------------
// I-BERT IntSoftmax for MI455X (gfx1250), rows of 1024 f32, softmax on last axis.
//
// Memory-bound kernel (536 MB traffic -> ~23us floor @ 23.3 TB/s); integer math
// trimmed to ~15 VALU/elem in the exp pass so compute stays under the HBM floor.
// CDNA5 paths used:
//   - global_load_async_to_lds_b128 (ASYNCcnt) + s_wait_asynccnt to stage and
//     double-buffer 2 rows per wave (compute row0 while row1 is in flight)
//   - wave32 shuffle (ds-swizzle) reductions for row max / row sum
// Wave32 layout: 1 row / wave, 32 elems / lane as 8 float4 chunks (chunk id =
// j*32+lane -> coalesced 16B/lane global access, conflict-free ds_load_b128).
// Row pipeline and async-issue stage are kept as real loops (#pragma unroll 1)
// so the row worker exists exactly once statically (small I-cache footprint).
// ---------------------------------------------------------------------------

#define WPB 4    // waves per block (block = 128 threads)
#define RPW 2    // rows per wave (async double buffer)
#define NROW 1024
#define CPL 8    // float4 chunks per lane = 1024 / (32 lanes * 4)

static __device__ __forceinline__ int imax2(int a, int b) { return a > b ? a : b; }
static __device__ __forceinline__ int imin2(int a, int b) { return a < b ? a : b; }

struct Uni {
    float    inv_sf;
    int      ux_max;   // CONST * |x0_int|
    uint32_t magic;    // (2^20 / u0) + 1  -> q = (ux*magic)>>20 (valid u0 <= 186)
    int      u0;       // |floor(-ln2/sf)|
    int      b_int;    // floor((COEF1/COEF0)/sf)
    int      c_int;    // floor((1/COEF0)/sf^2)
    uint32_t nm;       // fixed-point mantissa of new_scale (ACC=23), pre-shifted
    int      tsm1;     // normalized so tsm1 >= 0; total right shift = tsm1+1+q
};

// One row of 1024: max-subtract int exp, QuantAct(16) requant, shift-normalize.
// Single unified shift path: nm/tsm1 are pre-normalized so tsm1 >= 0, and after
// the one 64-bit shift the value provably fits 32 bits (<= ~65534, since
// COEF0*COEF2 == 1 makes the scaling self-normalizing).
static __device__ __forceinline__ void process_row(const float* __restrict__ rbuf,
                                                   float* __restrict__ orow,
                                                   int lane, const Uni& U)
{
    int xi[32];
    int mx = -2147483647 - 1;

    // ---- pass A: recover integer logits (round-to-nearest), row max
#pragma unroll
    for (int j = 0; j < CPL; ++j) {
        const float4 v = *(const float4*)(rbuf + (j * 32 + lane) * 4);
        int a0 = __float2int_rn(v.x * U.inv_sf);
        int a1 = __float2int_rn(v.y * U.inv_sf);
        int a2 = __float2int_rn(v.z * U.inv_sf);
        int a3 = __float2int_rn(v.w * U.inv_sf);
        xi[4 * j + 0] = a0; xi[4 * j + 1] = a1;
        xi[4 * j + 2] = a2; xi[4 * j + 3] = a3;
        mx = imax2(mx, imax2(imax2(a0, a1), imax2(a2, a3)));
    }
#pragma unroll
    for (int off = 16; off >= 1; off >>= 1)
        mx = imax2(mx, __shfl_xor(mx, off, 32));

    // ---- pass B: int_exp + FixedPointMul requant, exact integer row sum
    int isum = 0;
#pragma unroll
    for (int t = 0; t < 32; ++t) {
        int ux = imin2(mx - xi[t], U.ux_max);                  // ux = max - x_int, clamped
        uint32_t q = __umul24((uint32_t)ux, U.magic) >> 20;    // floor(ux / u0)
        int rem = ux - (int)__umul24(q, (uint32_t)U.u0);       // r = -rem, rem in [0, u0)
        int z = U.c_int + __mul24(rem, rem - U.b_int);         // (r+b)*r + c  (> 0)
        uint64_t tt = (uint64_t)(uint32_t)z * (uint64_t)U.nm;  // exp_int * nm, shift-folded
        int sh = imin2(U.tsm1 + (int)q, 63);                   // ts-1 >= 0 (UB guard at 63)
        uint32_t s1 = (uint32_t)(tt >> sh);                    // fits 32 bits
        int eq = imin2((int)((s1 + 1u) >> 1), 32767);          // round-half-away + clamp
        xi[t] = eq;                                            // reuse register array
        isum += eq;                                            // <= 1024*32767 < 2^25, exact
    }
#pragma unroll
    for (int off = 16; off >= 1; off >>= 1)
        isum += __shfl_xor(isum, off, 32);

    // ---- pass C: shift-normalize, emit fixed-point output (scale 1/256)
    const float factor = floorf(4294967296.0f / (float)isum);  // floor(2^32 / sum)
    const float fscale = factor * 0x1p-24f;                    // exact pow2 fold of /2^(32-8)
#pragma unroll
    for (int j = 0; j < CPL; ++j) {
        float4 r;
        r.x = floorf((float)xi[4 * j + 0] * fscale) * 0x1p-8f;
        r.y = floorf((float)xi[4 * j + 1] * fscale) * 0x1p-8f;
        r.z = floorf((float)xi[4 * j + 2] * fscale) * 0x1p-8f;
        r.w = floorf((float)xi[4 * j + 3] * fscale) * 0x1p-8f;
        *(float4*)(orow + (j * 32 + lane) * 4) = r;
    }
}

__global__ __launch_bounds__(WPB * 32)
void intsoftmax_gfx1250(const float* __restrict__ x,
                        const float* __restrict__ sfp,
                        float* __restrict__ out,
                        int nrows)
{
    __shared__ __align__(16) float sh[WPB * RPW * NROW];   // 32 KB
    const int lane = threadIdx.x & 31;
    const int wave = threadIdx.x >> 5;

    // ---- uniforms derived from the runtime scaling factor (sf = 0.1 in harness)
    const float sf = sfp[0];
    Uni U;
    U.inv_sf = 1.0f / sf;
    const int x0_int = (int)floorf(-0.6931f / sf);             // -7 for sf=0.1
    U.u0     = -x0_int;
    U.ux_max = 30 * U.u0;
    U.magic  = (1048576u / (uint32_t)U.u0) + 1u;
    U.b_int  = (int)floorf((0.96963238f / 0.35815147f) / sf);  // 27
    U.c_int  = (int)floorf((1.0f / 0.35815147f) / (sf * sf));  // 279
    const float act_sf = 1.0f / 32767.0f;
    const float exp_sf = (0.35815147f * sf * sf) * (1.0f / 1073741824.0f);
    const float nscale = exp_sf / act_sf;
    int efr;
    const float mant = frexpf(nscale, &efr);
    uint32_t nm  = (uint32_t)lroundf(mant * 8388608.0f);       // ACC = 23
    int      tsm1 = (23 - efr - 30) - 1;                       // = 15 for sf=0.1
    // Normalize so a single right-shift path suffices (only triggers for sf>13):
    const int k = imin2(imax2(-tsm1, 0), 8);
    U.nm   = nm << k;
    U.tsm1 = imax2(tsm1, 0);

    const long long row0 = ((long long)blockIdx.x * WPB + wave) * RPW;
    float* wbase = &sh[wave * (RPW * NROW)];
    const bool v1 = (row0 + 1 < (long long)nrows);

    // ---- stage rows global -> LDS with CDNA5 async copies (one static copy;
    //      the same instruction offset applies to both global src and LDS dst).
#pragma unroll 1
    for (int rr = 0; rr < RPW; ++rr) {
        if (row0 + rr >= (long long)nrows) break;
        const float* src_ = x + (size_t)(row0 + rr) * NROW;
        uint32_t la_ = (uint32_t)(uintptr_t)(wbase + rr * NROW) + (uint32_t)(lane * 16);
        uint64_t ga_ = (uint64_t)(uintptr_t)src_ + (uint64_t)(lane * 16);
        asm volatile("global_load_async_to_lds_b128 %0, %1, off offset:0"    :: "v"(la_), "v"(ga_) : "memory");
        asm volatile("global_load_async_to_lds_b128 %0, %1, off offset:512"  :: "v"(la_), "v"(ga_) : "memory");
        asm volatile("global_load_async_to_lds_b128 %0, %1, off offset:1024" :: "v"(la_), "v"(ga_) : "memory");
        asm volatile("global_load_async_to_lds_b128 %0, %1, off offset:1536" :: "v"(la_), "v"(ga_) : "memory");
        asm volatile("global_load_async_to_lds_b128 %0, %1, off offset:2048" :: "v"(la_), "v"(ga_) : "memory");
        asm volatile("global_load_async_to_lds_b128 %0, %1, off offset:2560" :: "v"(la_), "v"(ga_) : "memory");
        asm volatile("global_load_async_to_lds_b128 %0, %1, off offset:3072" :: "v"(la_), "v"(ga_) : "memory");
        asm volatile("global_load_async_to_lds_b128 %0, %1, off offset:3584" :: "v"(la_), "v"(ga_) : "memory");
    }

    // ---- software pipeline: wait for row0's 8 copies only (row1 in flight),
    //      compute row0, then drain and compute row1. One static row worker.
#pragma unroll 1
    for (int rr = 0; rr < RPW; ++rr) {
        const long long row = row0 + rr;
        if (row >= (long long)nrows) break;
        if (rr == 0 && v1) asm volatile("s_wait_asynccnt 0x8" ::: "memory");
        else               asm volatile("s_wait_asynccnt 0x0" ::: "memory");
        process_row(wbase + rr * NROW, out + (size_t)row * NROW, lane, U);
    }
}

extern "C" void kernel_launch(void* const* d_in, const int* in_sizes, int n_in,
                              void* d_out, int out_size, void* d_ws, size_t ws_size,
                              hipStream_t stream) {
    (void)n_in; (void)out_size; (void)d_ws; (void)ws_size;
    const float* x   = (const float*)d_in[0];
    const float* sfp = (const float*)d_in[1];
    float* out = (float*)d_out;

    const int nrows = in_sizes[0] / NROW;                 // 65536 for the reference shape
    const int rows_per_block = WPB * RPW;                 // 8
    const int grid = (nrows + rows_per_block - 1) / rows_per_block;

    intsoftmax_gfx1250<<<dim3(grid), dim3(WPB * 32), 0, stream>>>(x, sfp, out, nrows);
}